// GatedGCN_45054206935080
// MI455X (gfx1250) — compile-verified
//
#include <hip/hip_runtime.h>
#include <math.h>

#define D_DIM   128
#define EPS_BN  1e-5f
#define EPS_AGG 1e-6f

typedef float v2f __attribute__((ext_vector_type(2)));
typedef float v8f __attribute__((ext_vector_type(8)));

// ---------------------------------------------------------------------------
// GEMM: Y[M,128] = X[M,128] @ W[128,128] + bias[128]   (full fp32 via WMMA)
// 256 threads = 8 waves; block covers 256 rows; each wave a 32x128 strip
// (two 16-row tiles -> every B fragment feeds 2 WMMAs).
// W staged in LDS in fragment-ready layout: Wlds[(k>>1)*256 + col*2 + (k&1)]
// so a B fragment (W[k][col], W[k+1][col]) is a single aligned b64 LDS load.
// ---------------------------------------------------------------------------
__global__ __launch_bounds__(256)
void gemm128_bias(const float* __restrict__ X, const float* __restrict__ W,
                  const float* __restrict__ bias, float* __restrict__ Y, int M)
{
    __shared__ float Wlds[64 * 256];   // 64 KB

    // stage + swizzle W into LDS (coalesced global reads)
    for (int i = threadIdx.x; i < D_DIM * D_DIM; i += 256) {
        const int k = i >> 7, n = i & 127;
        Wlds[((k >> 1) << 8) + (n << 1) + (k & 1)] = W[i];
    }
    __syncthreads();

    const int wave  = threadIdx.x >> 5;
    const int lane  = threadIdx.x & 31;
    const int r     = lane & 15;          // row-in-tile for A, col-in-tile for B/C
    const int khalf = (lane >> 4) << 1;   // lanes 0-15 hold K+0..1, lanes 16-31 K+2..3

    const int rowBase = blockIdx.x * 256 + wave * 32;
    int row0 = rowBase + r;       if (row0 > M - 1) row0 = M - 1;   // clamp tail loads
    int row1 = rowBase + 16 + r;  if (row1 > M - 1) row1 = M - 1;
    const float* x0 = X + (size_t)row0 * D_DIM;
    const float* x1 = X + (size_t)row1 * D_DIM;

    v8f acc0[8] = {};   // 8 column tiles x 2 row tiles of 16x16 f32
    v8f acc1[8] = {};

    for (int kk = 0; kk < D_DIM; kk += 4) {
        // A fragments: 16x4 f32 layout (ISA 7.12.2), one aligned b64 load each
        const v2f a0 = *reinterpret_cast<const v2f*>(x0 + kk + khalf);
        const v2f a1 = *reinterpret_cast<const v2f*>(x1 + kk + khalf);
        const float* brow = &Wlds[(((kk + khalf) >> 1) << 8) + (r << 1)];
        #pragma unroll
        for (int c = 0; c < 8; ++c) {
            const v2f b = *reinterpret_cast<const v2f*>(brow + (c << 5));
            acc0[c] = __builtin_amdgcn_wmma_f32_16x16x4_f32(
                false, a0, false, b, (short)0, acc0[c], false, false);
            acc1[c] = __builtin_amdgcn_wmma_f32_16x16x4_f32(
                false, a1, false, b, (short)0, acc1[c], false, false);
        }
    }

    // C/D layout: VGPR v = row v (lanes 0-15) / row v+8 (lanes 16-31), col = lane&15
    const int rowOff = (lane >> 4) << 3;
    #pragma unroll
    for (int c = 0; c < 8; ++c) {
        const int   col = c * 16 + r;
        const float bv  = bias[col];
        #pragma unroll
        for (int v = 0; v < 8; ++v) {
            const int ra = rowBase + rowOff + v;
            if (ra < M) Y[(size_t)ra * D_DIM + col] = acc0[c][v] + bv;
            const int rb = ra + 16;
            if (rb < M) Y[(size_t)rb * D_DIM + col] = acc1[c][v] + bv;
        }
    }
}

// ---------------------------------------------------------------------------
// Edge pre-activation + per-channel BN statistics.
// thread = channel (0..127); each block walks `edgesPerBlock` edges.
// e_pre holds B3e on entry, overwritten with B1h[src]+B2h[dst]+B3e.
// ---------------------------------------------------------------------------
__global__ __launch_bounds__(128)
void edge_pre_bnstats(const float* __restrict__ B1h, const float* __restrict__ B2h,
                      float* __restrict__ e_pre,
                      const int* __restrict__ src, const int* __restrict__ dst,
                      float* __restrict__ bn_sum, float* __restrict__ bn_sumsq,
                      int E, int edgesPerBlock)
{
    const int c  = threadIdx.x;
    const int e0 = blockIdx.x * edgesPerBlock;
    const int e1 = min(e0 + edgesPerBlock, E);
    float s = 0.f, s2 = 0.f;
    for (int ed = e0; ed < e1; ++ed) {
        const int si = src[ed], di = dst[ed];
        float pre = e_pre[(size_t)ed * D_DIM + c]
                  + B1h[(size_t)si * D_DIM + c]
                  + B2h[(size_t)di * D_DIM + c];
        e_pre[(size_t)ed * D_DIM + c] = pre;
        s += pre; s2 += pre * pre;
    }
    atomicAdd(&bn_sum[c], s);
    atomicAdd(&bn_sumsq[c], s2);
}

// ---------------------------------------------------------------------------
// BN finalize: scale = gamma*rsqrt(var+eps), shift = beta - mean*scale
// ---------------------------------------------------------------------------
__global__ __launch_bounds__(128)
void bn_finalize(const float* __restrict__ sum, const float* __restrict__ sumsq,
                 const float* __restrict__ gamma, const float* __restrict__ beta,
                 float* __restrict__ scale, float* __restrict__ shift, float count)
{
    const int c  = threadIdx.x;
    const float mu  = sum[c] / count;
    const float var = sumsq[c] / count - mu * mu;   // biased variance (jnp.var)
    const float a   = gamma[c] * rsqrtf(var + EPS_BN);
    scale[c] = a;
    shift[c] = beta[c] - mu * a;
}

// ---------------------------------------------------------------------------
// sigma = sigmoid(relu(BN(e_pre)) + e); 4-way segment-sum scatter (atomics).
// ---------------------------------------------------------------------------
__global__ __launch_bounds__(128)
void edge_sigma_scatter(const float* __restrict__ e_pre, const float* __restrict__ e_in,
                        const float* __restrict__ A2h, const float* __restrict__ A3h,
                        const int* __restrict__ src, const int* __restrict__ dst,
                        const float* __restrict__ scale, const float* __restrict__ shift,
                        float* __restrict__ num_fwd, float* __restrict__ den_fwd,
                        float* __restrict__ num_bwd, float* __restrict__ den_bwd,
                        int E, int edgesPerBlock)
{
    const int   c = threadIdx.x;
    const float a = scale[c], b = shift[c];
    const int  e0 = blockIdx.x * edgesPerBlock;
    const int  e1 = min(e0 + edgesPerBlock, E);
    for (int ed = e0; ed < e1; ++ed) {
        const int si = src[ed], di = dst[ed];
        float act = a * e_pre[(size_t)ed * D_DIM + c] + b;
        act = fmaxf(act, 0.f) + e_in[(size_t)ed * D_DIM + c];
        const float sg = 1.0f / (1.0f + __expf(-act));
        atomicAdd(&num_fwd[(size_t)di * D_DIM + c], sg * A2h[(size_t)si * D_DIM + c]);
        atomicAdd(&den_fwd[(size_t)di * D_DIM + c], sg);
        atomicAdd(&num_bwd[(size_t)si * D_DIM + c], sg * A3h[(size_t)di * D_DIM + c]);
        atomicAdd(&den_bwd[(size_t)si * D_DIM + c], sg);
    }
}

// ---------------------------------------------------------------------------
// h_new = A1h + num_fwd/(den_fwd+eps) + num_bwd/(den_bwd+eps); BN stats.
// h_new written in place over A1h.
// ---------------------------------------------------------------------------
__global__ __launch_bounds__(128)
void node_update_bnstats(float* __restrict__ h_new,
                         const float* __restrict__ num_fwd, const float* __restrict__ den_fwd,
                         const float* __restrict__ num_bwd, const float* __restrict__ den_bwd,
                         float* __restrict__ bn_sum, float* __restrict__ bn_sumsq,
                         int Nn, int nodesPerBlock)
{
    const int c  = threadIdx.x;
    const int n0 = blockIdx.x * nodesPerBlock;
    const int n1 = min(n0 + nodesPerBlock, Nn);
    float s = 0.f, s2 = 0.f;
    for (int n = n0; n < n1; ++n) {
        const size_t idx = (size_t)n * D_DIM + c;
        float v = h_new[idx]
                + num_fwd[idx] / (den_fwd[idx] + EPS_AGG)
                + num_bwd[idx] / (den_bwd[idx] + EPS_AGG);
        h_new[idx] = v;
        s += v; s2 += v * v;
    }
    atomicAdd(&bn_sum[c], s);
    atomicAdd(&bn_sumsq[c], s2);
}

// ---------------------------------------------------------------------------
// h_out = relu(BN(h_new)) + h
// ---------------------------------------------------------------------------
__global__ __launch_bounds__(128)
void node_out_kernel(const float* __restrict__ h_new, const float* __restrict__ h_in,
                     const float* __restrict__ scale, const float* __restrict__ shift,
                     float* __restrict__ out, int Nn, int nodesPerBlock)
{
    const int   c = threadIdx.x;
    const float a = scale[c], b = shift[c];
    const int  n0 = blockIdx.x * nodesPerBlock;
    const int  n1 = min(n0 + nodesPerBlock, Nn);
    for (int n = n0; n < n1; ++n) {
        const size_t idx = (size_t)n * D_DIM + c;
        out[idx] = fmaxf(a * h_new[idx] + b, 0.f) + h_in[idx];
    }
}

// ---------------------------------------------------------------------------
extern "C" void kernel_launch(void* const* d_in, const int* in_sizes, int n_in,
                              void* d_out, int out_size, void* d_ws, size_t ws_size,
                              hipStream_t stream)
{
    const float* h    = (const float*)d_in[0];
    const float* e    = (const float*)d_in[1];
    const float* W_A1 = (const float*)d_in[2];  const float* b_A1 = (const float*)d_in[3];
    const float* W_A2 = (const float*)d_in[4];  const float* b_A2 = (const float*)d_in[5];
    const float* W_A3 = (const float*)d_in[6];  const float* b_A3 = (const float*)d_in[7];
    const float* W_B1 = (const float*)d_in[8];  const float* b_B1 = (const float*)d_in[9];
    const float* W_B2 = (const float*)d_in[10]; const float* b_B2 = (const float*)d_in[11];
    const float* W_B3 = (const float*)d_in[12]; const float* b_B3 = (const float*)d_in[13];
    const float* gamma_e = (const float*)d_in[14];
    const float* beta_e  = (const float*)d_in[15];
    const float* gamma_h = (const float*)d_in[16];
    const float* beta_h  = (const float*)d_in[17];
    const int*   src = (const int*)d_in[18];
    const int*   dst = (const int*)d_in[19];

    const int Nn = in_sizes[0] / D_DIM;   // 50000
    const int Ee = in_sizes[1] / D_DIM;   // 800000

    // workspace layout (floats)
    float* ws = (float*)d_ws;
    const size_t nd  = (size_t)Nn * D_DIM;
    const size_t edn = (size_t)Ee * D_DIM;
    float* A1h  = ws;              float* A2h  = A1h + nd;
    float* A3h  = A2h + nd;        float* B1h  = A3h + nd;
    float* B2h  = B1h + nd;        float* epre = B2h + nd;   // B3e -> e_pre
    float* num_fwd = epre + edn;   float* den_fwd = num_fwd + nd;
    float* num_bwd = den_fwd + nd; float* den_bwd = num_bwd + nd;
    float* stats   = den_bwd + nd; // [8][128]: sumE,sumsqE,scaleE,shiftE,sumH,sumsqH,scaleH,shiftH

    // zero aggregation buffers + statistics (single contiguous region)
    hipMemsetAsync(num_fwd, 0, (4 * nd + 8 * D_DIM) * sizeof(float), stream);

    const dim3 blk(256);
    const int gN = (Nn + 255) / 256;
    const int gE = (Ee + 255) / 256;
    gemm128_bias<<<gN, blk, 0, stream>>>(h, W_A1, b_A1, A1h, Nn);
    gemm128_bias<<<gN, blk, 0, stream>>>(h, W_A2, b_A2, A2h, Nn);
    gemm128_bias<<<gN, blk, 0, stream>>>(h, W_A3, b_A3, A3h, Nn);
    gemm128_bias<<<gN, blk, 0, stream>>>(h, W_B1, b_B1, B1h, Nn);
    gemm128_bias<<<gN, blk, 0, stream>>>(h, W_B2, b_B2, B2h, Nn);
    gemm128_bias<<<gE, blk, 0, stream>>>(e, W_B3, b_B3, epre, Ee);

    const int EPB = 64;
    edge_pre_bnstats<<<(Ee + EPB - 1) / EPB, 128, 0, stream>>>(
        B1h, B2h, epre, src, dst, stats + 0, stats + 128, Ee, EPB);
    bn_finalize<<<1, 128, 0, stream>>>(
        stats + 0, stats + 128, gamma_e, beta_e, stats + 256, stats + 384, (float)Ee);
    edge_sigma_scatter<<<(Ee + EPB - 1) / EPB, 128, 0, stream>>>(
        epre, e, A2h, A3h, src, dst, stats + 256, stats + 384,
        num_fwd, den_fwd, num_bwd, den_bwd, Ee, EPB);

    const int NPB = 64;
    node_update_bnstats<<<(Nn + NPB - 1) / NPB, 128, 0, stream>>>(
        A1h, num_fwd, den_fwd, num_bwd, den_bwd, stats + 512, stats + 640, Nn, NPB);
    bn_finalize<<<1, 128, 0, stream>>>(
        stats + 512, stats + 640, gamma_h, beta_h, stats + 768, stats + 896, (float)Nn);
    node_out_kernel<<<(Nn + NPB - 1) / NPB, 128, 0, stream>>>(
        A1h, h, stats + 768, stats + 896, (float*)d_out, Nn, NPB);
}